// TtMistralAttention_76519137345612
// MI455X (gfx1250) — compile-verified
//
#include <hip/hip_runtime.h>
#include <math.h>

// ---------------------------------------------------------------------------
// CDNA5 (gfx1250) WMMA fp32 path: V_WMMA_F32_16X16X4_F32
//   A: 16x4  f32 -> 2 VGPRs/lane  (lane m=lane&15 holds A[m][k+2g], A[m][k+2g+1], g=lane>>4)
//   B: 4x16  f32 -> 2 VGPRs/lane  (lane n=lane&15 holds B[2g][n],   B[2g+1][n])
//   C/D: 16x16 f32 -> 8 VGPRs     (VGPR r, lanes 0-15: M=r, N=lane; lanes 16-31: M=r+8)
// Plus: Tensor Data Mover staging of K tiles into LDS (tensor_load_to_lds,
// s_wait_tensorcnt) and global_prefetch_b8 on the streamed weight/V rows.
// wid is computed via readfirstlane so all per-wave loop bounds stay scalar
// (no EXEC-masked loops around the WMMAs).
// ---------------------------------------------------------------------------
typedef __attribute__((ext_vector_type(2))) float v2f;
typedef __attribute__((ext_vector_type(8))) float v8f;
typedef __attribute__((ext_vector_type(4))) unsigned int u32x4;
typedef __attribute__((ext_vector_type(8))) int i32x8;
typedef __attribute__((ext_vector_type(4))) int i32x4;

#define N_HEADS 32
#define N_KV 8
#define HEAD_DIM 128
#define DIM 4096
#define BATCH 32
#define KV_LEN 4096
#define QKV_COLS ((N_HEADS + 2 * N_KV) * HEAD_DIM) /* 6144 */
#define MAXL KV_LEN
#define KPITCH 132                 /* staged K row pitch (bank-conflict-free) */
#define KSTAGE_F (16 * KPITCH)     /* floats per staged 16x128 tile */

__device__ __forceinline__ v8f wmma_f32(v2f a, v2f b, v8f c) {
  return __builtin_amdgcn_wmma_f32_16x16x4_f32(
      /*neg_a=*/false, a, /*neg_b=*/false, b,
      /*c_mod=*/(short)0, c, /*reuse_a=*/false, /*reuse_b=*/false);
}

// Wave id as a provably-scalar (SGPR) value.
__device__ __forceinline__ int wave_id_scalar() {
  return __builtin_amdgcn_readfirstlane((int)threadIdx.x) >> 5;
}

// Low 32 bits of a generic shared-space address are the LDS byte offset
// (flat aperture mapping, ISA 10.2).
__device__ __forceinline__ unsigned int lds_offset(const void* p) {
  return (unsigned int)(uintptr_t)p;
}

// TDM: DMA one 16x128 f32 tile (row-major, row stride 128 f32) from global
// into LDS with 4-dword padding every 128 dwords (-> LDS pitch 132 f32).
// D# per ISA 8.3/8.4: group0 {count=1, lds_addr, global_addr, type=2},
// group1 {data_size=4B, pad_en, pad_interval=128dw, pad_amount=4dw,
//         tensor_dim0=128, tensor_dim1=16, tile_dim0=128, tile_dim1=16,
//         tensor_dim0_stride=128}. Groups 2/3/4 zero (2D tensor).
__device__ __forceinline__ void tdm_load_k_tile(unsigned int lds_byte_addr,
                                                const float* gsrc) {
  unsigned long long ga = (unsigned long long)(uintptr_t)gsrc;
  u32x4 g0;
  g0[0] = 1u;                                                    // count=1
  g0[1] = lds_byte_addr;                                         // lds_addr
  g0[2] = (unsigned int)ga;                                      // addr[31:0]
  g0[3] = (unsigned int)((ga >> 32) & 0x01FFFFFFu) | (2u << 30); // addr[56:32]|type=2
  i32x8 g1;
  g1[0] = (int)((2u << 16) | (1u << 20) | (6u << 22) | (3u << 25));
  g1[1] = (int)(((unsigned)HEAD_DIM & 0xFFFFu) << 16);  // tensor_dim0 lo16
  g1[2] = (int)(16u << 16);                             // dim0 hi | tensor_dim1 lo
  g1[3] = (int)((unsigned)HEAD_DIM << 16);              // dim1 hi | tile_dim0=128
  g1[4] = 16;                                           // tile_dim1=16, tile_dim2=0
  g1[5] = HEAD_DIM;                                     // tensor_dim0_stride
  g1[6] = 0;
  g1[7] = 0;
  i32x4 z4 = {0, 0, 0, 0};
  i32x8 z8 = {0, 0, 0, 0, 0, 0, 0, 0};
  __builtin_amdgcn_tensor_load_to_lds(g0, g1, z4, z4, z8, 0);
}

// ---------------------------------------------------------------------------
// C(32,N) = A(32,K) @ B(K,N), row-major. One 16-wide N-tile per BLOCK;
// the 4 waves of the block split K and reduce partials through LDS
// (fixed order -> deterministic). Each B element is read exactly once.
// grid = N/16 blocks, 128 threads (4 waves). K must be a multiple of 16.
// ---------------------------------------------------------------------------
__global__ void gemm32_wmma(const float* __restrict__ A,
                            const float* __restrict__ B,
                            float* __restrict__ C, int N, int K) {
  __shared__ float redbuf[3 * 512];
  const int lane = threadIdx.x & 31;
  const int wid = wave_id_scalar();     // SGPR -> scalar loop bounds below
  const int n0 = blockIdx.x * 16;
  const int m = lane & 15;
  const int g = lane >> 4;
  const int col = n0 + m;

  const int kslice = K >> 2;
  const int kb = wid * kslice;
  const int ke = kb + kslice;

  // Strength-reduced row pointers (no per-iteration 64-bit mul).
  const float* a0p = A + (size_t)m * K + 2 * g;
  const float* a1p = A + (size_t)(m + 16) * K + 2 * g;
  const float* bp0 = B + (size_t)(kb + 2 * g) * N + col;
  const float* bp1 = bp0 + N;
  const float* pf0 = bp0 + (size_t)128 * N;
  const float* pf1 = bp1 + (size_t)128 * N;
  const size_t bstep = (size_t)4 * N;

  v8f c0 = {};
  v8f c1 = {};
  int k = kb;
  const int kmain = ke - 128; // prefetch-covered portion (no guard branch)
  for (; k < kmain; k += 4) {
    __builtin_prefetch(pf0, 0, 1);
    __builtin_prefetch(pf1, 0, 1);
    v2f a0 = *(const v2f*)(a0p + k);
    v2f a1 = *(const v2f*)(a1p + k);
    v2f b = {*bp0, *bp1};
    c0 = wmma_f32(a0, b, c0);
    c1 = wmma_f32(a1, b, c1);
    bp0 += bstep; bp1 += bstep; pf0 += bstep; pf1 += bstep;
  }
  for (; k < ke; k += 4) { // tail without prefetch
    v2f a0 = *(const v2f*)(a0p + k);
    v2f a1 = *(const v2f*)(a1p + k);
    v2f b = {*bp0, *bp1};
    c0 = wmma_f32(a0, b, c0);
    c1 = wmma_f32(a1, b, c1);
    bp0 += bstep; bp1 += bstep;
  }

  // Cross-wave K reduction through LDS (waves 1..3 -> wave 0).
  if (wid > 0) {
    float* dst = redbuf + (wid - 1) * 512;
    for (int r = 0; r < 8; ++r) {
      dst[r * 32 + lane] = c0[r];
      dst[256 + r * 32 + lane] = c1[r];
    }
  }
  __syncthreads();
  if (wid == 0) {
    for (int w = 0; w < 3; ++w) {
      const float* src = redbuf + w * 512;
      for (int r = 0; r < 8; ++r) {
        c0[r] += src[r * 32 + lane];
        c1[r] += src[256 + r * 32 + lane];
      }
    }
    for (int r = 0; r < 8; ++r) {
      int row = r + 8 * g;
      C[(size_t)row * N + col] = c0[r];
      C[(size_t)(row + 16) * N + col] = c1[r];
    }
  }
}

// ---------------------------------------------------------------------------
// Rotary: out(rows,128) = gathered_rows(rows,128) @ rot(128,128)
// ---------------------------------------------------------------------------
__global__ void rot_gemm(const float* __restrict__ xqkv,
                         const float* __restrict__ rot,
                         float* __restrict__ out, int col_off, int hpb) {
  const int lane = threadIdx.x & 31;
  const int mt = blockIdx.x >> 3;
  const int n0 = (blockIdx.x & 7) * 16;
  const int m = lane & 15;
  const int g = lane >> 4;

  const int row_a = mt * 16 + m;
  const int b = row_a / hpb;
  const int h = row_a % hpb;
  const float* arow = xqkv + (size_t)b * QKV_COLS + col_off + (size_t)h * HEAD_DIM;

  v8f c = {};
  for (int k = 0; k < HEAD_DIM; k += 4) {
    v2f a = *(const v2f*)(arow + k + 2 * g);
    v2f bb;
    bb.x = rot[(size_t)(k + 2 * g) * HEAD_DIM + n0 + m];
    bb.y = rot[(size_t)(k + 2 * g + 1) * HEAD_DIM + n0 + m];
    c = wmma_f32(a, bb, c);
  }
  for (int r = 0; r < 8; ++r) {
    int row = mt * 16 + r + 8 * g;
    out[(size_t)row * HEAD_DIM + n0 + m] = c[r];
  }
}

// v slice copy out of xqkv (no rotary on V)
__global__ void v_copy(const float* __restrict__ xqkv, float* __restrict__ v_new) {
  int i = blockIdx.x * blockDim.x + threadIdx.x;
  if (i < BATCH * N_KV * HEAD_DIM) {
    int b = i / (N_KV * HEAD_DIM);
    int rem = i % (N_KV * HEAD_DIM);
    v_new[i] = xqkv[(size_t)b * QKV_COLS + (N_HEADS + N_KV) * HEAD_DIM + rem];
  }
}

// ---------------------------------------------------------------------------
// Attention: one workgroup (8 waves) per (b,kv) pair.
//  Phase 1a: full K tiles TDM-staged into LDS (double-buffered per wave),
//            QK^T via WMMA -> scores in LDS
//  Phase 1b: tail tile(s) containing row `cur` via direct spliced loads
//  Phase 2 : per-query softmax (unnormalized probs in LDS, 1/sum saved)
//  Phase 3 : PV via WMMA; pointer-bumped pure-cache main loop + spliced tail
// Row `cur` of K/V comes from workspace (inputs are never mutated).
// ---------------------------------------------------------------------------
__global__ void attn_kernel(const float* __restrict__ q_rot,
                            const float* __restrict__ k_new,
                            const float* __restrict__ v_new,
                            const float* __restrict__ cache_k,
                            const float* __restrict__ cache_v,
                            const int* __restrict__ pos_ptr,
                            float* __restrict__ attn_out) {
  extern __shared__ float smem[];
  float* scores = smem;                    // 5 * MAXL (row 4 = zeros pad)
  float* qs = scores + 5 * MAXL;           // 16 * 128 (rows 4..15 zero pad)
  float* kn = qs + 16 * HEAD_DIM;          // 128
  float* vn = kn + HEAD_DIM;               // 128
  float* red = vn + HEAD_DIM;              // 16
  float* kstage = red + 16;                // 8 waves * 2 bufs * KSTAGE_F

  const int b = blockIdx.x >> 3;
  const int kv = blockIdx.x & 7;
  const int tid = threadIdx.x;
  const int lane = tid & 31;
  const int wid = wave_id_scalar();        // SGPR -> scalar tile loops below
  const int m = lane & 15;
  const int g = lane >> 4;

  const int cur = *pos_ptr;
  const int L = cur + 1;
  const int ntiles = (L + 15) >> 4;
  const float scale = 0.08838834764831845f; // 1/sqrt(128)

  const float* Kc = cache_k + ((size_t)b * N_KV + kv) * (size_t)KV_LEN * HEAD_DIM;
  const float* Vc = cache_v + ((size_t)b * N_KV + kv) * (size_t)KV_LEN * HEAD_DIM;

  // Preloads: q group zero-padded to 16 rows, new K/V rows, zero scores row 4.
  for (int i = tid; i < 16 * HEAD_DIM; i += blockDim.x)
    qs[i] = (i < 4 * HEAD_DIM)
                ? q_rot[((size_t)b * N_HEADS + kv * 4) * HEAD_DIM + i]
                : 0.f;
  if (tid < HEAD_DIM) {
    kn[tid] = k_new[((size_t)b * N_KV + kv) * HEAD_DIM + tid];
    vn[tid] = v_new[((size_t)b * N_KV + kv) * HEAD_DIM + tid];
  }
  for (int i = tid; i < MAXL; i += blockDim.x) scores[4 * MAXL + i] = 0.f;
  __syncthreads();

  // ----- Phase 1a: TDM-staged pure-cache tiles (all 16 rows < cur) -----
  const int nfull = cur >> 4;
  {
    float* buf0 = kstage + (size_t)wid * 2 * KSTAGE_F;
    float* buf1 = buf0 + KSTAGE_F;
    int buf = 0;
    if (wid < nfull)
      tdm_load_k_tile(lds_offset(buf0), Kc + (size_t)wid * 16 * HEAD_DIM);
    for (int t = wid; t < nfull; t += 8) {
      const int tn = t + 8;
      if (tn < nfull) {
        tdm_load_k_tile(lds_offset(buf ? buf0 : buf1),
                        Kc + (size_t)tn * 16 * HEAD_DIM);
        __builtin_amdgcn_s_wait_tensorcnt(1); // tile t landed; tn in flight
      } else {
        __builtin_amdgcn_s_wait_tensorcnt(0);
      }
      const float* kt = buf ? buf1 : buf0;
      v8f c = {};
      for (int k = 0; k < HEAD_DIM; k += 4) {
        v2f a = *(const v2f*)(qs + m * HEAD_DIM + k + 2 * g);
        v2f bb = *(const v2f*)(kt + m * KPITCH + k + 2 * g);
        c = wmma_f32(a, bb, c);
      }
      if (g == 0)
        for (int r = 0; r < 4; ++r)
          scores[r * MAXL + t * 16 + m] = c[r] * scale;
      buf ^= 1;
    }
  }

  // ----- Phase 1b: tail tile(s) with the spliced `cur` row -----
  for (int t = nfull + wid; t < ntiles; t += 8) {
    const int s = t * 16 + m;
    const float* krow = (s == cur) ? kn : (s < cur ? Kc + (size_t)s * HEAD_DIM : nullptr);
    v8f c = {};
    for (int k = 0; k < HEAD_DIM; k += 4) {
      v2f a = *(const v2f*)(qs + m * HEAD_DIM + k + 2 * g);
      v2f bb;
      bb.x = krow ? krow[k + 2 * g] : 0.f;
      bb.y = krow ? krow[k + 2 * g + 1] : 0.f;
      c = wmma_f32(a, bb, c);
    }
    if (g == 0)
      for (int r = 0; r < 4; ++r)
        scores[r * MAXL + t * 16 + m] = (t * 16 + m < L) ? c[r] * scale : -1e30f;
  }
  __syncthreads();

  // ----- Phase 2: softmax (wave q handles query q) -----
  if (wid < 4) {
    float* sc = scores + wid * MAXL;
    float mx = -1e30f;
    for (int i = lane; i < L; i += 32) mx = fmaxf(mx, sc[i]);
    for (int o = 16; o; o >>= 1) mx = fmaxf(mx, __shfl_xor(mx, o, 32));
    float sum = 0.f;
    for (int i = lane; i < L; i += 32) {
      float p = __expf(sc[i] - mx);
      sc[i] = p;
      sum += p;
    }
    for (int o = 16; o; o >>= 1) sum += __shfl_xor(sum, o, 32);
    for (int i = L + lane; i < ntiles * 16; i += 32) sc[i] = 0.f; // zero tail
    if (lane == 0) red[wid] = 1.0f / sum;
  }
  __syncthreads();

  // ----- Phase 3: PV (wave wid handles d-tile wid) -----
  const int d0 = wid * 16;
  const int rm = (m < 4) ? m : 4; // rows >=4 read the zero row
  const float* sr = scores + (size_t)rm * MAXL;
  v8f acc = {};
  const int send = ntiles * 16;
  const int s_pure = (cur > 3) ? ((cur - 3) & ~3) : 0;
  int s_pf = s_pure; // prefetch-covered portion of the pure loop
  if (send >= 64) { if (s_pf > send - 64) s_pf = send - 64; } else s_pf = 0;
  if (s_pf < 0) s_pf = 0;

  const float* vp0 = Vc + (size_t)(2 * g) * HEAD_DIM + d0 + m;
  const float* vp1 = vp0 + HEAD_DIM;
  const float* vpf = vp0 + (size_t)64 * HEAD_DIM;
  int s0 = 0;
  for (; s0 < s_pf; s0 += 4) { // pure-cache, prefetched
    __builtin_prefetch(vpf, 0, 1);
    v2f a = *(const v2f*)(sr + s0 + 2 * g);
    v2f bb = {*vp0, *vp1};
    acc = wmma_f32(a, bb, acc);
    vp0 += 4 * HEAD_DIM; vp1 += 4 * HEAD_DIM; vpf += 4 * HEAD_DIM;
  }
  for (; s0 < s_pure; s0 += 4) { // pure-cache, no prefetch
    v2f a = *(const v2f*)(sr + s0 + 2 * g);
    v2f bb = {*vp0, *vp1};
    acc = wmma_f32(a, bb, acc);
    vp0 += 4 * HEAD_DIM; vp1 += 4 * HEAD_DIM;
  }
  for (; s0 < send; s0 += 4) { // spliced tail
    v2f a = *(const v2f*)(sr + s0 + 2 * g);
    const int sb0 = s0 + 2 * g, sb1 = sb0 + 1;
    const float* vr0 = (sb0 == cur) ? vn : (sb0 < cur ? Vc + (size_t)sb0 * HEAD_DIM : nullptr);
    const float* vr1 = (sb1 == cur) ? vn : (sb1 < cur ? Vc + (size_t)sb1 * HEAD_DIM : nullptr);
    v2f bb;
    bb.x = vr0 ? vr0[d0 + m] : 0.f;
    bb.y = vr1 ? vr1[d0 + m] : 0.f;
    acc = wmma_f32(a, bb, acc);
  }
  if (g == 0) {
    for (int r = 0; r < 4; ++r) {
      float val = acc[r] * red[r];
      attn_out[((size_t)b * N_HEADS + kv * 4 + r) * HEAD_DIM + d0 + m] = val;
    }
  }
}

// ---------------------------------------------------------------------------
extern "C" void kernel_launch(void* const* d_in, const int* in_sizes, int n_in,
                              void* d_out, int out_size, void* d_ws, size_t ws_size,
                              hipStream_t stream) {
  (void)in_sizes; (void)n_in; (void)out_size; (void)ws_size;
  const float* x = (const float*)d_in[0];        // (32, 4096)
  const float* wqkv = (const float*)d_in[1];     // (4096, 6144)
  const float* wo = (const float*)d_in[2];       // (4096, 4096)
  const float* rot = (const float*)d_in[3];      // (128, 128)
  const float* cache_k = (const float*)d_in[4];  // (32, 8, 4096, 128)
  const float* cache_v = (const float*)d_in[5];  // (32, 8, 4096, 128)
  const int* pos = (const int*)d_in[6];          // scalar (device)
  float* out = (float*)d_out;                    // (32, 4096)

  float* ws = (float*)d_ws;
  float* xqkv_ws = ws;                                   // 32*6144
  float* q_rot = xqkv_ws + BATCH * QKV_COLS;             // 32*32*128
  float* k_new = q_rot + BATCH * N_HEADS * HEAD_DIM;     // 32*8*128
  float* v_new = k_new + BATCH * N_KV * HEAD_DIM;        // 32*8*128
  float* attn = v_new + BATCH * N_KV * HEAD_DIM;         // 32*4096

  // 1) QKV projection (one N-tile per block, 4 waves split K)
  gemm32_wmma<<<QKV_COLS / 16, 128, 0, stream>>>(x, wqkv, xqkv_ws, QKV_COLS, DIM);
  // 2) Rotary on q and k; copy v
  rot_gemm<<<(BATCH * N_HEADS / 16) * 8, 32, 0, stream>>>(xqkv_ws, rot, q_rot, 0, N_HEADS);
  rot_gemm<<<(BATCH * N_KV / 16) * 8, 32, 0, stream>>>(xqkv_ws, rot, k_new,
                                                       N_HEADS * HEAD_DIM, N_KV);
  v_copy<<<(BATCH * N_KV * HEAD_DIM + 255) / 256, 256, 0, stream>>>(xqkv_ws, v_new);
  // 3) Attention (one WG per (b,kv)); ~221KB dynamic LDS (WGP has 320KB)
  size_t shmem = (size_t)(5 * MAXL + 16 * HEAD_DIM + 2 * HEAD_DIM + 16 +
                          8 * 2 * KSTAGE_F) * sizeof(float);
  (void)hipFuncSetAttribute((const void*)attn_kernel,
                            hipFuncAttributeMaxDynamicSharedMemorySize, (int)shmem);
  attn_kernel<<<BATCH * N_KV, 256, shmem, stream>>>(q_rot, k_new, v_new, cache_k,
                                                    cache_v, pos, attn);
  // 4) Output projection
  gemm32_wmma<<<DIM / 16, 128, 0, stream>>>(attn, wo, out, DIM, DIM);
}